// layer_to_tt_tensor_learnable_60473139528505
// MI455X (gfx1250) — compile-verified
//
#include <hip/hip_runtime.h>

// TT-SVD sweep for MI455X (gfx1250): one workgroup per (batch,channel) sample,
// all intermediates LDS-resident, fp32 WMMA (v_wmma_f32_16x16x4_f32) for the
// dominant Gram-matrix GEMM and subspace-iteration multiplies.

typedef __attribute__((ext_vector_type(2))) float v2f;
typedef __attribute__((ext_vector_type(8))) float v8f;

#define TT_B    512
#define TT_C    3
#define NSAMP   (TT_B * TT_C)
#define HW      128
#define NEL     (HW * HW)       // 16384 floats per sample
#define OUTLEN  2176
#define NITER   3               // subspace-iteration sweeps
#define JSWEEPS 10              // Jacobi sweeps for small symmetric eig

// ---- LDS partition offsets (floats) ----
#define OFF_X    0              // 16384: A0 (8x2048) -> reused as G1 (128x128)
#define OFF_T1   16384          // 16384: tens1 = A1 (128x128)
#define OFF_Q    32768          // 2048 : Q (128x16)
#define OFF_Z    34816          // 2048 : Z (128x16) (also scratch for G0 partials)
#define OFF_RED  36864          // 160  : block reductions
#define OFF_M8   37024          // 64   : 8x8 symmetric scratch
#define OFF_V8   37088          // 64   : 8x8 eigenvectors
#define OFF_M16  37152          // 256  : 16x16 Rayleigh-Ritz matrix
#define OFF_V16  37408          // 256  : 16x16 eigenvectors
#define OFF_EV   37664          // 16   : eigenvalues
#define OFF_SV   37680          // 16   : singular values
#define OFF_U1   37696          // 1024 : U1 (128x8)
#define OFF_SVT  38720          // 1024 : S*V^T (8x128)
#define OFF_VV   39744          // 1024 : vv (128x8)
#define OFF_T2   40768          // 1024 : tens2 (8x128) == A2 (128x8) flat
#define OFF_U2   41792          // 1024 : U2 (128x8)
#define LDS_FLOATS 42816        // 171,264 bytes < 320 KB/WGP

__device__ __forceinline__ v8f wmma_f32_k4(v2f a, v2f b, v8f c) {
  // D(16x16,f32) += A(16x4,f32) * B(4x16,f32)
  return __builtin_amdgcn_wmma_f32_16x16x4_f32(
      /*neg_a=*/false, a, /*neg_b=*/false, b,
      /*c_mod=*/(short)0, c, /*reuse_a=*/false, /*reuse_b=*/false);
}

// Block-wide sum over first 128 lanes' contributions. All 256 threads must call.
__device__ __forceinline__ float block_sum_128(float* red, int tid, float v) {
  if (tid < 128) red[tid] = v;
  __syncthreads();
  if (tid == 0) {
    float s = 0.f;
    for (int i = 0; i < 128; ++i) s += red[i];
    red[128] = s;
  }
  __syncthreads();
  float r = red[128];
  __syncthreads();
  return r;
}

// Serial cyclic Jacobi eigensolver for small symmetric n x n (thread 0 only).
__device__ void jacobi_sym(float* A, float* V, int n, int sweeps) {
  for (int i = 0; i < n * n; ++i) V[i] = 0.f;
  for (int i = 0; i < n; ++i) V[i * n + i] = 1.f;
  for (int s = 0; s < sweeps; ++s) {
    for (int p = 0; p < n - 1; ++p) {
      for (int q = p + 1; q < n; ++q) {
        float apq = A[p * n + q];
        if (fabsf(apq) < 1e-12f) continue;
        float app = A[p * n + p], aqq = A[q * n + q];
        float tau = (aqq - app) / (2.f * apq);
        float t = (tau >= 0.f ? 1.f : -1.f) / (fabsf(tau) + sqrtf(1.f + tau * tau));
        float cc = rsqrtf(1.f + t * t);
        float ss = t * cc;
        for (int k = 0; k < n; ++k) {
          float akp = A[k * n + p], akq = A[k * n + q];
          A[k * n + p] = cc * akp - ss * akq;
          A[k * n + q] = ss * akp + cc * akq;
        }
        for (int k = 0; k < n; ++k) {
          float apk = A[p * n + k], aqk = A[q * n + k];
          A[p * n + k] = cc * apk - ss * aqk;
          A[q * n + k] = ss * apk + cc * aqk;
        }
        for (int k = 0; k < n; ++k) {
          float vkp = V[k * n + p], vkq = V[k * n + q];
          V[k * n + p] = cc * vkp - ss * vkq;
          V[k * n + q] = ss * vkp + cc * vkq;
        }
      }
    }
  }
}

__device__ void sort_eigs_desc(const float* A, float* V, float* eig, int n) {
  for (int i = 0; i < n; ++i) eig[i] = A[i * n + i];
  for (int i = 0; i < n; ++i) {
    int m = i;
    for (int j = i + 1; j < n; ++j)
      if (eig[j] > eig[m]) m = j;
    if (m != i) {
      float t = eig[i]; eig[i] = eig[m]; eig[m] = t;
      for (int k = 0; k < n; ++k) {
        float t2 = V[k * n + i]; V[k * n + i] = V[k * n + m]; V[k * n + m] = t2;
      }
    }
  }
}

// Z(128x16) = G(128x128) * Q(128x16), one 16x16 output tile per wave via WMMA.
__device__ __forceinline__ void gq_multiply(const float* G, const float* Qm,
                                            float* Zm, int wave, int l, int half) {
  const float* Grow = G + (wave * 16 + l) * HW;  // A-matrix rows
  v8f acc = {};
  for (int k0 = 0; k0 < HW; k0 += 4) {
    v2f a = *(const v2f*)(Grow + k0 + 2 * half);   // K = k0+2h, k0+2h+1
    v2f bb;
    bb.x = Qm[(k0 + 2 * half) * 16 + l];           // B[K][N=l]
    bb.y = Qm[(k0 + 2 * half + 1) * 16 + l];
    acc = wmma_f32_k4(a, bb, acc);
  }
#pragma unroll
  for (int r = 0; r < 8; ++r)
    Zm[(wave * 16 + 8 * half + r) * 16 + l] = acc[r];
}

__global__ __launch_bounds__(256)
void tt_svd_kernel(const float* __restrict__ x,
                   const float* __restrict__ w_s,
                   const float* __restrict__ w_u,
                   const float* __restrict__ w_v,
                   float* __restrict__ out) {
  extern __shared__ float smem[];
  float* X   = smem + OFF_X;
  float* T1  = smem + OFF_T1;
  float* Q   = smem + OFF_Q;
  float* Z   = smem + OFF_Z;
  float* red = smem + OFF_RED;
  float* M8  = smem + OFF_M8;
  float* V8  = smem + OFF_V8;
  float* M16 = smem + OFF_M16;
  float* V16 = smem + OFF_V16;
  float* EV  = smem + OFF_EV;
  float* SV  = smem + OFF_SV;
  float* U1  = smem + OFF_U1;
  float* SVt = smem + OFF_SVT;
  float* VV  = smem + OFF_VV;
  float* T2  = smem + OFF_T2;
  float* U2  = smem + OFF_U2;

  const int tid  = threadIdx.x;
  const int lane = tid & 31;
  const int wave = tid >> 5;
  const int half = lane >> 4;
  const int l    = lane & 15;

  const int samp = blockIdx.x;          // samp = b*C + c
  const int bidx = samp / TT_C;
  const int cidx = samp % TT_C;

  const float* xs = x + (size_t)samp * NEL;
  float* outp = out + (size_t)samp * OUTLEN;

  // ---- Stage in sample: 16384 floats, float4 streaming loads ----
  {
    const float4* xv = (const float4*)xs;
    float4* Xv = (float4*)X;
    for (int i = tid; i < NEL / 4; i += 256) Xv[i] = xv[i];
  }
  __syncthreads();

  // ================= Step 0: SVD of A0 = X viewed 8 x 2048 =================
  // G0 = A0 * A0^T (8x8): 64 entries, 4 partial threads each
  {
    int e = tid >> 2, part = tid & 3;
    int r = e >> 3, c2 = e & 7;
    const float* ar = X + r * 2048 + part * 512;
    const float* ac = X + c2 * 2048 + part * 512;
    float p = 0.f;
    for (int k = 0; k < 512; ++k) p += ar[k] * ac[k];
    Z[tid] = p;                                    // Z as scratch here
  }
  __syncthreads();
  if (tid < 64)
    M8[tid] = Z[tid * 4] + Z[tid * 4 + 1] + Z[tid * 4 + 2] + Z[tid * 4 + 3];
  __syncthreads();
  if (tid == 0) { jacobi_sym(M8, V8, 8, JSWEEPS); sort_eigs_desc(M8, V8, EV, 8); }
  __syncthreads();

  // core1 = U0 (8x8 row-major)
  if (tid < 64) outp[tid] = V8[tid];

  // tens1 = U0^T * A0  (8x2048) == A1 (128x128 row-major)
  for (int o = tid; o < NEL; o += 256) {
    int r = o >> 11, k = o & 2047;
    float acc = 0.f;
#pragma unroll
    for (int m = 0; m < 8; ++m) acc += V8[m * 8 + r] * X[m * 2048 + k];
    T1[o] = acc;
  }
  __syncthreads();

  // ================= Step 1: top-8 SVD of A1 (128x128) =================
  // G1 = A1 * A1^T via WMMA f32 16x16x4; 64 tiles, 8 per wave; store into X.
  for (int t = 0; t < 8; ++t) {
    int tile = wave * 8 + t;
    int I = tile >> 3, J = tile & 7;
    const float* Arow = T1 + (I * 16 + l) * HW;
    const float* Brow = T1 + (J * 16 + l) * HW;   // B[K][N=l] = A1[16J+l][K]
    v8f acc = {};
    for (int k0 = 0; k0 < HW; k0 += 4) {
      v2f a  = *(const v2f*)(Arow + k0 + 2 * half);
      v2f bb = *(const v2f*)(Brow + k0 + 2 * half);
      acc = wmma_f32_k4(a, bb, acc);
    }
    int col = J * 16 + l;
#pragma unroll
    for (int r = 0; r < 8; ++r)
      X[(I * 16 + 8 * half + r) * HW + col] = acc[r];
  }
  __syncthreads();

  // Subspace iteration on G1 with 16-column block (top-8 kept after RR).
  for (int i = tid; i < 2048; i += 256) {
    int rr = i >> 4, jj = i & 15;
    unsigned h = (unsigned)(rr * 1973 + jj * 9277 + 12345);
    h ^= h >> 13; h *= 2654435761u; h ^= h >> 16;
    Q[i] = ((float)(h & 0xFFFFu) / 32768.0f) - 1.0f;
  }
  __syncthreads();

  for (int it = 0; it < NITER; ++it) {
    gq_multiply(X, Q, Z, wave, l, half);          // Z = G1 * Q (WMMA)
    __syncthreads();
    // Modified Gram-Schmidt: Z columns -> orthonormal Q
    for (int j = 0; j < 16; ++j) {
      for (int i2 = 0; i2 < j; ++i2) {
        float v = (tid < 128) ? Z[tid * 16 + j] * Q[tid * 16 + i2] : 0.f;
        float proj = block_sum_128(red, tid, v);
        if (tid < 128) Z[tid * 16 + j] -= proj * Q[tid * 16 + i2];
        __syncthreads();
      }
      float v = (tid < 128) ? Z[tid * 16 + j] * Z[tid * 16 + j] : 0.f;
      float nrm = block_sum_128(red, tid, v);
      float inv = rsqrtf(nrm + 1e-20f);
      if (tid < 128) Q[tid * 16 + j] = Z[tid * 16 + j] * inv;
      __syncthreads();
    }
  }

  // Rayleigh-Ritz: Z = G1*Q, T = Q^T Z, eig(T) -> rotate Q
  gq_multiply(X, Q, Z, wave, l, half);
  __syncthreads();
  {
    int m = tid >> 4, n = tid & 15;
    float acc = 0.f;
    for (int k = 0; k < HW; ++k) acc += Q[k * 16 + m] * Z[k * 16 + n];
    M16[m * 16 + n] = acc;
  }
  __syncthreads();
  if (tid == 0) { jacobi_sym(M16, V16, 16, JSWEEPS); sort_eigs_desc(M16, V16, EV, 16); }
  __syncthreads();
  if (tid < 8) SV[tid] = sqrtf(fmaxf(EV[tid], 0.f));
  __syncthreads();

  // U1 = Q * V16[:, :8]  (128x8)
  for (int o = tid; o < 1024; o += 256) {
    int rr = o >> 3, j = o & 7;
    float acc = 0.f;
#pragma unroll
    for (int k = 0; k < 16; ++k) acc += Q[rr * 16 + k] * V16[k * 16 + j];
    U1[o] = acc;
  }
  __syncthreads();

  const float* wu1 = w_u + (size_t)(cidx * 3 + 1) * 64;
  const float* wv1 = w_v + (size_t)(cidx * 3 + 1) * 64;
  const float* ws1 = w_s + ((size_t)(bidx * TT_C + cidx) * 3 + 1) * 8;

  // core2 = U1 @ wu1  (128x8)
  for (int o = tid; o < 1024; o += 256) {
    int rr = o >> 3, j = o & 7;
    float acc = 0.f;
#pragma unroll
    for (int k = 0; k < 8; ++k) acc += U1[rr * 8 + k] * wu1[k * 8 + j];
    outp[64 + o] = acc;
  }

  // S*V^T = U1^T * A1  (8x128)
  for (int o = tid; o < 1024; o += 256) {
    int r = o >> 7, k = o & 127;
    float acc = 0.f;
    for (int m = 0; m < HW; ++m) acc += U1[m * 8 + r] * T1[m * HW + k];
    SVt[o] = acc;
  }
  __syncthreads();

  // vv = (A1^T U1 diag(1/s)) @ wv1  -> VV[n*8+j]
  for (int o = tid; o < 1024; o += 256) {
    int n = o >> 3, j = o & 7;
    float acc = 0.f;
#pragma unroll
    for (int r = 0; r < 8; ++r) {
      float invs = SV[r] > 1e-12f ? 1.0f / SV[r] : 0.f;
      acc += SVt[r * HW + n] * invs * wv1[r * 8 + j];
    }
    VV[o] = acc;
  }
  __syncthreads();

  // tens2[r][n] = (ws1[r]*s[r]) * vv[n][r]   (8x128 flat == A2 128x8)
  for (int o = tid; o < 1024; o += 256) {
    int r = o >> 7, n = o & 127;
    T2[o] = ws1[r] * SV[r] * VV[n * 8 + r];
  }
  __syncthreads();

  // ================= Step 2: SVD of A2 (128x8) =================
  // G2 = A2^T A2 (8x8)
  if (tid < 64) {
    int r = tid >> 3, c2 = tid & 7;
    float acc = 0.f;
    for (int k = 0; k < HW; ++k) acc += T2[k * 8 + r] * T2[k * 8 + c2];
    M8[tid] = acc;
  }
  __syncthreads();
  if (tid == 0) { jacobi_sym(M8, V8, 8, JSWEEPS); sort_eigs_desc(M8, V8, EV, 8); }
  __syncthreads();
  if (tid < 8) SV[tid] = sqrtf(fmaxf(EV[tid], 0.f));
  __syncthreads();

  // U2 = A2 * V2 * diag(1/s)  (128x8)
  for (int o = tid; o < 1024; o += 256) {
    int i = o >> 3, j = o & 7;
    float acc = 0.f;
#pragma unroll
    for (int k = 0; k < 8; ++k) acc += T2[i * 8 + k] * V8[k * 8 + j];
    float invs = SV[j] > 1e-12f ? 1.0f / SV[j] : 0.f;
    U2[o] = acc * invs;
  }
  __syncthreads();

  const float* wu2 = w_u + (size_t)(cidx * 3 + 2) * 64;
  const float* wv2 = w_v + (size_t)(cidx * 3 + 2) * 64;
  const float* ws2 = w_s + ((size_t)(bidx * TT_C + cidx) * 3 + 2) * 8;

  // core3 = U2 @ wu2  (128x8)
  for (int o = tid; o < 1024; o += 256) {
    int i = o >> 3, j = o & 7;
    float acc = 0.f;
#pragma unroll
    for (int k = 0; k < 8; ++k) acc += U2[i * 8 + k] * wu2[k * 8 + j];
    outp[64 + 1024 + o] = acc;
  }

  // core4: tens3[r][n] = (ws2[r]*s[r]) * (V2 @ wv2)[n][r]  (8x8)
  if (tid < 64) {
    int r = tid >> 3, n = tid & 7;
    float acc = 0.f;
#pragma unroll
    for (int k = 0; k < 8; ++k) acc += V8[n * 8 + k] * wv2[k * 8 + r];
    outp[64 + 1024 + 1024 + r * 8 + n] = ws2[r] * SV[r] * acc;
  }
}

extern "C" void kernel_launch(void* const* d_in, const int* in_sizes, int n_in,
                              void* d_out, int out_size, void* d_ws, size_t ws_size,
                              hipStream_t stream) {
  const float* x   = (const float*)d_in[0];
  const float* w_s = (const float*)d_in[1];
  const float* w_u = (const float*)d_in[2];
  const float* w_v = (const float*)d_in[3];
  float* out = (float*)d_out;

  dim3 grid(NSAMP);
  dim3 block(256);
  size_t shmem = (size_t)LDS_FLOATS * sizeof(float);
  hipLaunchKernelGGL(tt_svd_kernel, grid, block, shmem, stream,
                     x, w_s, w_u, w_v, out);
}